// MyModel_80771154968593
// MI455X (gfx1250) — compile-verified
//
#include <hip/hip_runtime.h>
#include <hip/hip_bf16.h>

// Problem constants (from the reference): B=32, S=4096, H=768, fp32.
#define BS      (32 * 4096)       // 131072 tokens
#define H       768
#define HHALF   (H / 2)           // 384: low-half lanes sweep H[0..383], high-half H[384..767]
#define KSTEPS  (H / 4)           // 192 WMMA K-steps of 4
#define PAIRS   (KSTEPS / 2)      // 96 loop iterations, 2 WMMAs per iteration
#define TPB     256               // 8 wave32s per block
#define TOKS_PER_BLOCK (8 * 16)   // each wave owns a 16-token tile
#define NBLOCKS (BS / TOKS_PER_BLOCK) // 1024

typedef __attribute__((ext_vector_type(2))) float v2f;
typedef __attribute__((ext_vector_type(4))) float v4f;
typedef __attribute__((ext_vector_type(8))) float v8f;

// D = A(16x4) * B(4x16) + C via V_WMMA_F32_16X16X4_F32.
//   A rows: M=0 -> w_start, M=1 -> w_end, M>=2 -> 0 (zero block in LDS).
//   B cols: N = token-in-tile.
// K remap per step i: K-positions {0,1} = H{2i,2i+1}, {2,3} = H{384+2i,384+2i+1}.
// This makes every lane's A and B streams byte-contiguous -> b128 loads feed 2 WMMAs.
// After 192 steps, lane n<16 holds start_logit in acc[0] (row M=0) and
// end_logit in acc[1] (row M=1).
__global__ __launch_bounds__(TPB)
void dual_gemv_wmma_kernel(const float* __restrict__ hs,
                           const float* __restrict__ w_start,
                           const float* __restrict__ w_end,
                           float* __restrict__ out) {
    // LDS: w_start verbatim [0..767], w_end verbatim [768..1535], zero block [1536..1539].
    __shared__ float lds[2 * H + 4];

    const int t = threadIdx.x;

    // ---- Stage weights into LDS (straight copy; the K remap needs no shuffle) ----
    if (t < H / 4) {                          // 192 threads, one float4 each
        ((float4*)lds)[t]         = ((const float4*)w_start)[t];
        ((float4*)lds)[H / 4 + t] = ((const float4*)w_end)[t];
    } else if (t == H / 4) {
        ((float4*)lds)[2 * (H / 4)] = make_float4(0.f, 0.f, 0.f, 0.f);  // zero block
    }
    __syncthreads();

    const int lane  = t & 31;       // wave32
    const int wave  = t >> 5;       // 0..7
    const int m     = lane & 15;    // A row (weight select) / B column (token)
    const int khalf = lane >> 4;    // 0: K-pos {0,1} (H low half), 1: K-pos {2,3} (H high half)

    // B stream: this lane's token row, starting at its H half; +16B per iteration.
    const long tok = (long)blockIdx.x * TOKS_PER_BLOCK + wave * 16 + m;
    const float* bp = hs + tok * H + khalf * HHALF;

    // A stream in LDS: w_m region + H-half offset; rows >=2 pin to the zero block.
    const bool valid  = (m < 2);
    int       fidx    = valid ? (m * H + khalf * HHALF) : (2 * H);
    const int fstride = valid ? 4 : 0;   // floats per iteration (16 B)

    v8f acc = {};

    #pragma unroll 4
    for (int i = 0; i < PAIRS; ++i) {
        v4f a4 = *(const v4f*)&lds[fidx];  // ds_load_b128     : A chunks for 2 steps
        v4f b4 = *(const v4f*)bp;          // global_load_b128 : hidden chunks for 2 steps
        v2f a0 = {a4.x, a4.y}, a1 = {a4.z, a4.w};
        v2f b0 = {b4.x, b4.y}, b1 = {b4.z, b4.w};
        // 8 args: (neg_a, A, neg_b, B, c_mod, C, reuse_a, reuse_b)
        acc = __builtin_amdgcn_wmma_f32_16x16x4_f32(
            false, a0, false, b0, (short)0, acc, false, false);
        acc = __builtin_amdgcn_wmma_f32_16x16x4_f32(
            false, a1, false, b1, (short)0, acc, false, false);
        fidx += fstride;
        bp   += 4;
    }

    // D layout: lanes 0-15, VGPR0 = row M=0 (start), VGPR1 = row M=1 (end).
    if (lane < 16) {
        out[tok]      = acc[0];   // start_logits, flat [B*S]
        out[BS + tok] = acc[1];   // end_logits,   flat [B*S]
    }
}

extern "C" void kernel_launch(void* const* d_in, const int* in_sizes, int n_in,
                              void* d_out, int out_size, void* d_ws, size_t ws_size,
                              hipStream_t stream) {
    const float* hs      = (const float*)d_in[0];  // [B,S,H] f32
    const float* w_start = (const float*)d_in[1];  // [H] f32
    const float* w_end   = (const float*)d_in[2];  // [H] f32
    float* out           = (float*)d_out;          // [2, B*S] f32 (tuple concat)

    dim3 grid(NBLOCKS), block(TPB);
    dual_gemv_wmma_kernel<<<grid, block, 0, stream>>>(hs, w_start, w_end, out);
}